// Attention_15659450761267
// MI455X (gfx1250) — compile-verified
//
#include <hip/hip_runtime.h>

// ---------------- problem constants (from reference) ----------------
#define S_SEQ      64
#define H_Q        32
#define HKV        8
#define GQ         4          // H_Q / HKV query heads per kv head
#define HD         128        // head dim
#define BS         16         // cache block size (== one WMMA key tile)
#define MAX_BLOCKS 128
#define SCALE      0.08838834764831845f
#define NEG_BIG    (-1.0e30f)

#define NW         4          // waves per workgroup (wave32)
#define NTHREADS   (NW * 32)

typedef float f32x2 __attribute__((ext_vector_type(2)));
typedef float f32x8 __attribute__((ext_vector_type(8)));

// ---------------- LDS layout (float indices) ----------------
// per wave: double-buffered K/V tiles + P relayout scratch
#define TILE_SZ   (BS * HD)                    // 2048 fl (8KB) per K or V tile
#define WAVE_LDS  (4 * TILE_SZ + 16 * 16)      // 2 bufs x (K+V) + P = 8448 fl
#define POFF      (4 * TILE_SZ)                // P scratch offset within wave region
#define LDSO_OFF  (NW * WAVE_LDS)              // per-wave unnormalized O [NW][GQ][HD]
#define LDSM_OFF  (LDSO_OFF + NW * GQ * HD)    // per-wave running max  [NW][GQ]
#define LDSL_OFF  (LDSM_OFF + NW * GQ)         // per-wave running sum  [NW][GQ]
#define SMEM_FLOATS (LDSL_OFF + NW * GQ)

#define WAIT_ASYNC() asm volatile("s_wait_asynccnt 0x0" ::: "memory")

__device__ __forceinline__ void async_copy16(unsigned lds_addr, const float* gsrc) {
  // GLOBAL_LOAD_ASYNC_TO_LDS_B128, GV mode: per-lane 16B memory -> LDS, ASYNCcnt-tracked
  unsigned long long ga = (unsigned long long)(size_t)gsrc;
  asm volatile("global_load_async_to_lds_b128 %0, %1, off"
               :: "v"(lds_addr), "v"(ga) : "memory");
}

__global__ __launch_bounds__(NTHREADS) void paged_attn_decode_wmma(
    const float* __restrict__ q,        // [S, H, D]
    const float* __restrict__ knew,     // [S, HKV, D]
    const float* __restrict__ vnew,     // [S, HKV, D]
    const float* __restrict__ k_cache,  // [NUM_BLOCKS, BS, HKV, D]
    const float* __restrict__ v_cache,  // [NUM_BLOCKS, BS, HKV, D]
    const int*   __restrict__ block_tables, // [S, MAX_BLOCKS]
    const int*   __restrict__ context_lens, // [S]
    float*       __restrict__ out)      // [S, H, D]
{
  extern __shared__ float smem[];

  const int kv   = blockIdx.x;          // kv head
  const int s    = blockIdx.y;          // sequence
  const int tid  = threadIdx.x;
  const int w    = tid >> 5;            // wave id within WG
  const int lane = tid & 31;
  const int l16  = lane & 15;
  const int hsel = (lane >= 16) ? 2 : 0;   // K-offset selector for A/B 16x16x4 frags
  const int rofs = (lane >= 16) ? 8 : 0;   // row offset for C/D layout

  const int ctx    = context_lens[s];
  const int ntiles = (ctx + BS - 1) / BS;

  const int widx = w * WAVE_LDS;                         // wave LDS region (float idx)
  const unsigned wave_lds_byte = (unsigned)(size_t)(smem + widx);  // LDS byte base for async

  // ---- preload Q as 16x16x4 A-fragments: row m = l16, rows >= GQ are zero pad ----
  f32x2 qreg[32];
  {
    const float* qrow = q + ((size_t)(s * H_Q + kv * GQ + l16)) * HD;
    const f32x2 zz = {0.f, 0.f};
    #pragma unroll
    for (int st = 0; st < 32; ++st)
      qreg[st] = (l16 < GQ) ? *(const f32x2*)(qrow + st * 4 + hsel) : zz;
  }

  const float* kq = knew + ((size_t)(s * HKV + kv)) * HD;   // fresh token K row
  const float* vq = vnew + ((size_t)(s * HKV + kv)) * HD;   // fresh token V row

  // ---- async-stage one 16-key K/V tile into buffer `bufo` (float-index offset) ----
  auto stage = [&](int t, int bufo) {
    const int blk = block_tables[s * MAX_BLOCKS + t];
    const float* kb = k_cache + ((size_t)blk * BS * HKV + kv) * HD;
    const float* vb = v_cache + ((size_t)blk * BS * HKV + kv) * HD;
    if (t + NW < ntiles) {  // warm L2 one double-buffer depth further out
      const int blkn = block_tables[s * MAX_BLOCKS + t + NW];
      const float* kbn = k_cache + ((size_t)blkn * BS * HKV + kv) * HD;
      const float* vbn = v_cache + ((size_t)blkn * BS * HKV + kv) * HD;
      __builtin_prefetch(kbn + (size_t)(lane >> 1) * HKV * HD + (lane & 1) * 64, 0, 0);
      __builtin_prefetch(vbn + (size_t)(lane >> 1) * HKV * HD + (lane & 1) * 64, 0, 0);
    }
    const unsigned kbyte = wave_lds_byte + (unsigned)(bufo + lane * 4) * 4u;
    const unsigned vbyte = kbyte + (unsigned)TILE_SZ * 4u;
    #pragma unroll
    for (int i = 0; i < BS; ++i) {
      const int pos = t * BS + i;                     // wave-uniform
      const float* kr = (pos == ctx - 1) ? kq : (kb + (size_t)i * HKV * HD);
      const float* vr = (pos == ctx - 1) ? vq : (vb + (size_t)i * HKV * HD);
      async_copy16(kbyte + (unsigned)(i * HD) * 4u, kr + lane * 4);
      async_copy16(vbyte + (unsigned)(i * HD) * 4u, vr + lane * 4);
    }
  };

  // ---- flash-decode state (padded 16 rows; only rows 0..GQ-1 are real) ----
  f32x8 acc[8];
  const f32x8 zero8 = {0.f,0.f,0.f,0.f,0.f,0.f,0.f,0.f};
  #pragma unroll
  for (int c = 0; c < 8; ++c) acc[c] = zero8;
  float m_r[8], l_r[8];
  #pragma unroll
  for (int j = 0; j < 8; ++j) { m_r[j] = NEG_BIG; l_r[j] = 0.f; }

  if (w < ntiles) stage(w, 0);                 // prologue stage (wave-uniform cond)

  int par = 0;
  for (int t = w; t < ntiles; t += NW, par ^= 1) {
    WAIT_ASYNC();                              // tile t resident in buffer `par`
    if (t + NW < ntiles) stage(t + NW, (par ^ 1) * 2 * TILE_SZ);   // overlap next tile

    // integer-offset buffer select: keeps smem[] accesses in addrspace(3) -> ds_load
    const int kbase = widx + par * 2 * TILE_SZ;
    const int vbase = kbase + TILE_SZ;

    // ---- scores = Q (16x128) x K^T (128x16): 32 wmma, B-frags batched by 8 ----
    f32x8 sc = zero8;
    #pragma unroll
    for (int grp = 0; grp < 4; ++grp) {
      f32x2 bfr[8];
      #pragma unroll
      for (int u = 0; u < 8; ++u)
        bfr[u] = *(const f32x2*)&smem[kbase + l16 * HD + (grp * 8 + u) * 4 + hsel];
      #pragma unroll
      for (int u = 0; u < 8; ++u)
        sc = __builtin_amdgcn_wmma_f32_16x16x4_f32(
                false, qreg[grp * 8 + u], false, bfr[u], (short)0, sc, false, false);
    }

    // ---- mask + scale + online softmax (row j+rofs lives in reg j, 16-lane group) ----
    const int  kpos   = t * BS + l16;
    const bool kvalid = (kpos < ctx);
    float alpha[8];
    #pragma unroll
    for (int j = 0; j < 8; ++j) {
      float sj = kvalid ? (sc[j] * SCALE) : NEG_BIG;
      float mx = sj;
      #pragma unroll
      for (int off = 8; off >= 1; off >>= 1) mx = fmaxf(mx, __shfl_xor(mx, off, 32));
      const float mn = fmaxf(m_r[j], mx);
      const float al = __expf(m_r[j] - mn);
      const float p  = __expf(sj - mn);
      float rs = p;
      #pragma unroll
      for (int off = 8; off >= 1; off >>= 1) rs += __shfl_xor(rs, off, 32);
      l_r[j] = l_r[j] * al + rs;
      m_r[j] = mn;
      alpha[j] = al;
      smem[widx + POFF + (j + rofs) * 16 + l16] = p;   // D-format -> row-major scratch
    }
    #pragma unroll
    for (int c = 0; c < 8; ++c) {
      #pragma unroll
      for (int j = 0; j < 8; ++j) acc[c][j] *= alpha[j];
    }

    // ---- O += P (16x16) x V (16x128): 8 col-tiles x 4 wmma, loads batched ----
    f32x2 pa[4];
    #pragma unroll
    for (int st = 0; st < 4; ++st)
      pa[st] = *(const f32x2*)&smem[widx + POFF + l16 * 16 + st * 4 + hsel];
    #pragma unroll
    for (int c = 0; c < 8; ++c) {
      f32x2 bv[4];
      #pragma unroll
      for (int st = 0; st < 4; ++st) {
        bv[st].x = smem[vbase + (st * 4 + hsel)     * HD + c * 16 + l16];
        bv[st].y = smem[vbase + (st * 4 + hsel + 1) * HD + c * 16 + l16];
      }
      #pragma unroll
      for (int st = 0; st < 4; ++st)
        acc[c] = __builtin_amdgcn_wmma_f32_16x16x4_f32(
                    false, pa[st], false, bv[st], (short)0, acc[c], false, false);
    }
  }

  // ---- publish per-wave partials (only real rows 0..GQ-1, lanes 0..15 = half 0) ----
  if (lane < 16) {
    #pragma unroll
    for (int j = 0; j < GQ; ++j)
      #pragma unroll
      for (int c = 0; c < 8; ++c)
        smem[LDSO_OFF + (w * GQ + j) * HD + c * 16 + lane] = acc[c][j];
  }
  if (lane == 0) {
    #pragma unroll
    for (int j = 0; j < GQ; ++j) {
      smem[LDSM_OFF + w * GQ + j] = m_r[j];
      smem[LDSL_OFF + w * GQ + j] = l_r[j];
    }
  }
  __syncthreads();

  // ---- cross-wave log-sum-exp merge, write GQ x HD outputs ----
  #pragma unroll
  for (int rep = 0; rep < (GQ * HD) / NTHREADS; ++rep) {
    const int idx = tid + rep * NTHREADS;
    const int row = idx >> 7;        // HD == 128
    const int col = idx & (HD - 1);
    float M = NEG_BIG;
    #pragma unroll
    for (int w2 = 0; w2 < NW; ++w2) M = fmaxf(M, smem[LDSM_OFF + w2 * GQ + row]);
    float den = 0.f, num = 0.f;
    #pragma unroll
    for (int w2 = 0; w2 < NW; ++w2) {
      const float e = __expf(smem[LDSM_OFF + w2 * GQ + row] - M);
      den += e * smem[LDSL_OFF + w2 * GQ + row];
      num += e * smem[LDSO_OFF + (w2 * GQ + row) * HD + col];
    }
    out[((size_t)(s * H_Q + kv * GQ + row)) * HD + col] = num / den;
  }
}

extern "C" void kernel_launch(void* const* d_in, const int* in_sizes, int n_in,
                              void* d_out, int out_size, void* d_ws, size_t ws_size,
                              hipStream_t stream) {
  const float* q   = (const float*)d_in[0];
  const float* k   = (const float*)d_in[1];
  const float* v   = (const float*)d_in[2];
  const float* kc  = (const float*)d_in[3];
  const float* vc  = (const float*)d_in[4];
  const int*   bt  = (const int*)d_in[5];
  const int*   cl  = (const int*)d_in[6];
  // d_in[7] slot_mapping is implied by (context_lens, block_tables); not needed.
  float* out = (float*)d_out;

  const size_t smem = (size_t)SMEM_FLOATS * sizeof(float);   // ~140 KB (<= 320 KB/WGP)
  (void)hipFuncSetAttribute((const void*)paged_attn_decode_wmma,
                            hipFuncAttributeMaxDynamicSharedMemorySize, (int)smem);

  dim3 grid(HKV, S_SEQ);     // 512 workgroups: one per (kv head, sequence)
  dim3 block(NTHREADS);      // 4 wave32s split the context, double-buffered async staging
  hipLaunchKernelGGL(paged_attn_decode_wmma, grid, block, smem, stream,
                     q, k, v, kc, vc, bt, cl, out);
}